// CombinedLoss_4286377362142
// MI455X (gfx1250) — compile-verified
//
#include <hip/hip_runtime.h>
#include <hip/hip_bf16.h>
#include <stdint.h>

typedef float v2f __attribute__((ext_vector_type(2)));
typedef float v8f __attribute__((ext_vector_type(8)));

#define HH 256
#define INF_D 512          // h + w
#define THETA0 3.0f
#define THETA_INV 0.2f     // 1/5

// ---------------------------------------------------------------------------
// Kernel A: per-row horizontal L1 distance to nearest zero, for mask m (pos)
// and complement 1-m (neg).  One thread per (image,row).  g clamped to 512.
// ---------------------------------------------------------------------------
__global__ void edt_rows_kernel(const float* __restrict__ tgt,
                                uint16_t* __restrict__ gpos,
                                uint16_t* __restrict__ gneg,
                                int nrows) {
  int r = blockIdx.x * blockDim.x + threadIdx.x;   // r = img*256 + y
  if (r >= nrows) return;
  int base = r * HH;
  // forward sweep (left -> right)
  int fp = INF_D, fn = INF_D;
  for (int x = 0; x < HH; ++x) {
    float t = tgt[base + x];
    bool z = (t < 0.5f);           // zero pixel of m
    fp = z ? 0 : fp + 1;           // dist to nearest zero of m
    fn = z ? fn + 1 : 0;           // dist to nearest zero of (1-m)
    gpos[base + x] = (uint16_t)fp;
    gneg[base + x] = (uint16_t)fn;
  }
  // backward sweep (right -> left), combine, clamp to INF
  int bp = INF_D, bn = INF_D;
  for (int x = HH - 1; x >= 0; --x) {
    int fpv = gpos[base + x];
    int fnv = gneg[base + x];
    bool z  = (fpv == 0);          // zero of m
    bp = z ? 0 : bp + 1;
    bn = (fnv == 0) ? 0 : bn + 1;
    int gp = min(min(fpv, bp), INF_D);
    int gn = min(min(fnv, bn), INF_D);
    gpos[base + x] = (uint16_t)gp;
    gneg[base + x] = (uint16_t)gn;
  }
}

// ---------------------------------------------------------------------------
// Kernel B: exact vertical squared-EDT pass + boundary weight.
// One block = (image, 4 columns); thread y loops j with g^2 staged in LDS
// (broadcast reads, no bank conflicts).  D2[y,x] = min_j g2[j,x] + (y-j)^2.
// ---------------------------------------------------------------------------
#define COLS 4
__global__ void edt_cols_weight_kernel(const float* __restrict__ tgt,
                                       const uint16_t* __restrict__ gpos,
                                       const uint16_t* __restrict__ gneg,
                                       float* __restrict__ wts) {
  __shared__ float s_g2p[COLS][HH];
  __shared__ float s_g2n[COLS][HH];
  int img = blockIdx.x / (HH / COLS);
  int x0  = (blockIdx.x % (HH / COLS)) * COLS;
  int tid = threadIdx.x;

  for (int idx = tid; idx < COLS * HH; idx += blockDim.x) {
    int c = idx >> 8;        // column within group
    int j = idx & (HH - 1);  // row
    int gi = (img * HH + j) * HH + (x0 + c);
    float gp = (float)gpos[gi];
    float gn = (float)gneg[gi];
    s_g2p[c][j] = gp * gp;
    s_g2n[c][j] = gn * gn;
  }
  __syncthreads();

  int y = tid;
  float dp[COLS], dn[COLS];
#pragma unroll
  for (int c = 0; c < COLS; ++c) { dp[c] = 3.0e38f; dn[c] = 3.0e38f; }

  for (int j = 0; j < HH; ++j) {
    float dy  = (float)(y - j);
    float dy2 = dy * dy;
#pragma unroll
    for (int c = 0; c < COLS; ++c) {
      dp[c] = fminf(dp[c], s_g2p[c][j] + dy2);
      dn[c] = fminf(dn[c], s_g2n[c][j] + dy2);
    }
  }

#pragma unroll
  for (int c = 0; c < COLS; ++c) {
    int gi = (img * HH + y) * HH + (x0 + c);
    float t   = tgt[gi];
    float d2  = (t > 0.5f) ? dp[c] : dn[c];
    float ad  = sqrtf(d2);                       // |signed distance|
    // sigmoid((theta0 - |d|)/theta) = 1/(1+exp((|d|-theta0)/theta))
    wts[gi] = 1.0f / (1.0f + expf((ad - THETA0) * THETA_INV));
  }
}

// ---------------------------------------------------------------------------
// Kernel C: fused elementwise (BCE, sigmoid) + five global sums.
// Sums accumulated on the matrix pipe via V_WMMA_F32_16X16X4_F32:
// A = ones(16x4), B = 64 data values/wave, C/D = f32 accumulator.
// Every B slot maps to exactly one column sum; sum(D row0 over lanes) = 2*total.
// Uniform trip count -> EXEC all-ones at every WMMA (ISA requirement).
// Deterministic: fixed-order per-block partials, no float atomics.
// ---------------------------------------------------------------------------
__global__ void fused_reduce_kernel(const float* __restrict__ pred,
                                    const float* __restrict__ tgt,
                                    const float* __restrict__ wts,
                                    float* __restrict__ partials,
                                    int pairs, int iters) {
  const int gtid  = blockIdx.x * blockDim.x + threadIdx.x;
  const int gsize = gridDim.x * blockDim.x;

  v8f acc0 = {}, acc1 = {}, acc2 = {}, acc3 = {}, acc4 = {};
  v2f aones; aones.x = 1.0f; aones.y = 1.0f;

  for (int it = 0; it < iters; ++it) {           // uniform bound: no divergence
    int i  = gtid + it * gsize;
    int e0 = 2 * i, e1 = 2 * i + 1;

    float x0 = pred[e0], t0 = tgt[e0], w0 = wts[e0];
    float x1 = pred[e1], t1 = tgt[e1], w1 = wts[e1];

    float b0 = fmaxf(x0, 0.0f) - x0 * t0 + log1pf(expf(-fabsf(x0)));
    float b1 = fmaxf(x1, 0.0f) - x1 * t1 + log1pf(expf(-fabsf(x1)));
    float p0 = 1.0f / (1.0f + expf(-x0));
    float p1 = 1.0f / (1.0f + expf(-x1));

    v2f qb;  qb.x = b0;       qb.y = b1;        // bce
    v2f qp;  qp.x = p0;       qp.y = p1;        // sigmoid
    v2f qpt; qpt.x = p0 * t0; qpt.y = p1 * t1;  // p*t
    v2f qt;  qt.x = t0;       qt.y = t1;        // t
    v2f qbw; qbw.x = b0 * w0; qbw.y = b1 * w1;  // bce*w

    acc0 = __builtin_amdgcn_wmma_f32_16x16x4_f32(false, aones, false, qb,  (short)0, acc0, false, false);
    acc1 = __builtin_amdgcn_wmma_f32_16x16x4_f32(false, aones, false, qp,  (short)0, acc1, false, false);
    acc2 = __builtin_amdgcn_wmma_f32_16x16x4_f32(false, aones, false, qpt, (short)0, acc2, false, false);
    acc3 = __builtin_amdgcn_wmma_f32_16x16x4_f32(false, aones, false, qt,  (short)0, acc3, false, false);
    acc4 = __builtin_amdgcn_wmma_f32_16x16x4_f32(false, aones, false, qbw, (short)0, acc4, false, false);
  }

  // scalar tail (pairs not covered by the uniform loop; 0 for 16x256x256)
  float tl0 = 0.f, tl1 = 0.f, tl2 = 0.f, tl3 = 0.f, tl4 = 0.f;
  for (int i = gtid + iters * gsize; i < pairs; i += gsize) {
    for (int e = 2 * i; e < 2 * i + 2; ++e) {
      float x = pred[e], t = tgt[e], w = wts[e];
      float b = fmaxf(x, 0.0f) - x * t + log1pf(expf(-fabsf(x)));
      float p = 1.0f / (1.0f + expf(-x));
      tl0 += b; tl1 += p; tl2 += p * t; tl3 += t; tl4 += b * w;
    }
  }

  // per-lane value: D rows duplicate (row0 == row8), so lanes hold 2x coverage
  float v0 = acc0[0] * 0.5f + tl0;
  float v1 = acc1[0] * 0.5f + tl1;
  float v2 = acc2[0] * 0.5f + tl2;
  float v3 = acc3[0] * 0.5f + tl3;
  float v4 = acc4[0] * 0.5f + tl4;

  for (int off = 16; off > 0; off >>= 1) {       // wave32 tree reduce
    v0 += __shfl_xor(v0, off);
    v1 += __shfl_xor(v1, off);
    v2 += __shfl_xor(v2, off);
    v3 += __shfl_xor(v3, off);
    v4 += __shfl_xor(v4, off);
  }

  __shared__ float s_part[8][5];                 // 256 threads = 8 wave32
  int wave = threadIdx.x >> 5;
  int lane = threadIdx.x & 31;
  if (lane == 0) {
    s_part[wave][0] = v0; s_part[wave][1] = v1; s_part[wave][2] = v2;
    s_part[wave][3] = v3; s_part[wave][4] = v4;
  }
  __syncthreads();
  if (threadIdx.x == 0) {
    int nw = blockDim.x >> 5;
    float s0 = 0.f, s1 = 0.f, s2 = 0.f, s3 = 0.f, s4 = 0.f;
    for (int w = 0; w < nw; ++w) {
      s0 += s_part[w][0]; s1 += s_part[w][1]; s2 += s_part[w][2];
      s3 += s_part[w][3]; s4 += s_part[w][4];
    }
    float* p = partials + blockIdx.x * 5;
    p[0] = s0; p[1] = s1; p[2] = s2; p[3] = s3; p[4] = s4;
  }
}

// ---------------------------------------------------------------------------
// Kernel D: fold per-block partials (fixed order) and emit the 4 outputs.
// ---------------------------------------------------------------------------
__global__ void finalize_kernel(const float* __restrict__ partials,
                                float* __restrict__ out,
                                int nblocks, int n_elems) {
  __shared__ float sums[5];
  int tid = threadIdx.x;
  if (tid < 5) {
    float s = 0.0f;
    for (int b = 0; b < nblocks; ++b) s += partials[b * 5 + tid];
    sums[tid] = s;
  }
  __syncthreads();
  if (tid == 0) {
    float N = (float)n_elems;
    float bce_loss  = sums[0] / N;
    float dice      = (2.0f * sums[2] + 1.0f) / (sums[1] + sums[3] + 1.0f);
    float dice_loss = 1.0f - dice;
    float bnd_loss  = sums[4] / N;
    out[0] = bce_loss + dice_loss + bnd_loss;
    out[1] = bce_loss;
    out[2] = dice_loss;
    out[3] = bnd_loss;
  }
}

// ---------------------------------------------------------------------------
extern "C" void kernel_launch(void* const* d_in, const int* in_sizes, int n_in,
                              void* d_out, int out_size, void* d_ws, size_t ws_size,
                              hipStream_t stream) {
  const float* pred = (const float*)d_in[0];
  const float* tgt  = (const float*)d_in[1];
  float* out = (float*)d_out;

  int n    = in_sizes[0];            // 16*1*256*256 = 1048576
  int nimg = n / (HH * HH);
  int nrows = nimg * HH;

  // workspace layout
  uint16_t* gpos = (uint16_t*)d_ws;                         // nimg*65536 u16
  uint16_t* gneg = gpos + (size_t)nimg * HH * HH;           // nimg*65536 u16
  float*    wts  = (float*)(gneg + (size_t)nimg * HH * HH); // nimg*65536 f32
  float*    partials = wts + (size_t)nimg * HH * HH;        // NB_C*5 f32

  // A: horizontal distance scans (one thread per row)
  edt_rows_kernel<<<nimg, 256, 0, stream>>>(tgt, gpos, gneg, nrows);

  // B: vertical exact EDT + boundary weights (block = image x 4 columns)
  edt_cols_weight_kernel<<<nimg * (HH / COLS), HH, 0, stream>>>(tgt, gpos, gneg, wts);

  // C: fused elementwise + WMMA-accumulated global sums
  const int NB_C = 256, BS_C = 256;
  int gsize = NB_C * BS_C;
  int pairs = n / 2;
  int iters = pairs / gsize;
  fused_reduce_kernel<<<NB_C, BS_C, 0, stream>>>(pred, tgt, wts, partials, pairs, iters);

  // D: deterministic fold + final scalars
  finalize_kernel<<<1, 32, 0, stream>>>(partials, out, NB_C, n);
}